// MLPScorer_8589934807
// MI455X (gfx1250) — compile-verified
//
#include <hip/hip_runtime.h>

// Problem constants (B,T,S,D) = (4,512,512,512)
constexpr int Bc = 4;
constexpr int Tc = 512;
constexpr int Sc = 512;
constexpr int Dc = 512;

typedef __attribute__((ext_vector_type(2))) float v2f;
typedef __attribute__((ext_vector_type(8))) float v8f;

__device__ __forceinline__ float fast_tanh(float x) {
#if __has_builtin(__builtin_amdgcn_tanhf)
    return __builtin_amdgcn_tanhf(x);
#elif __has_builtin(__builtin_amdgcn_tanh_f32)
    return __builtin_amdgcn_tanh_f32(x);
#else
    float r;
    asm volatile("v_tanh_f32 %0, %1" : "=v"(r) : "v"(x));
    return r;
#endif
}

// Y[m,n] = sum_k X[m,k] * W[n,k] (+ bias[n])   i.e.  Y = X @ W^T + b
// One wave (32 lanes) computes a 16x16 tile via V_WMMA_F32_16X16X4_F32.
// A (16x4 f32) lane layout: lanes 0-15 hold M=lane, K=k0..k0+1 in v[0..1];
// lanes 16-31 hold M=lane-16, K=k0+2..k0+3.  B (4x16) is symmetric with
// N=lane&15, so both operands are contiguous float2 loads.
__global__ void __launch_bounds__(32)
proj_gemm_wmma(const float* __restrict__ X,   // [R, Dc] row-major
               const float* __restrict__ W,   // [Dc, Dc] row-major
               const float* __restrict__ bias,// [Dc] or nullptr
               float* __restrict__ Y) {       // [R, Dc]
    const int tile_n = blockIdx.x;          // Dc/16 tiles
    const int tile_m = blockIdx.y;          // R/16 tiles
    const int lane   = threadIdx.x;
    const int r      = lane & 15;
    const int half   = lane >> 4;           // 0: K pair {0,1}, 1: K pair {2,3}

    const float* arow = X + (size_t)(tile_m * 16 + r) * Dc + half * 2;
    const float* brow = W + (size_t)(tile_n * 16 + r) * Dc + half * 2;

    v8f acc = {};
#pragma unroll 8
    for (int k = 0; k < Dc; k += 4) {
        v2f a = *(const v2f*)(arow + k);
        v2f b = *(const v2f*)(brow + k);
        acc = __builtin_amdgcn_wmma_f32_16x16x4_f32(
            /*neg_a=*/false, a, /*neg_b=*/false, b,
            /*c_mod=*/(short)0, acc, /*reuse_a=*/false, /*reuse_b=*/false);
    }

    const int n = tile_n * 16 + r;
    const float badd = bias ? bias[n] : 0.0f;
    // C/D layout: VGPR i -> M = i + 8*half, N = lane&15
#pragma unroll
    for (int i = 0; i < 8; ++i) {
        const int m = tile_m * 16 + i + half * 8;
        Y[(size_t)m * Dc + n] = acc[i] + badd;
    }
}

// out[b,t,s] = sum_d v[d] * tanh(wq[b,t,d] + uh[b,s,d])
// Block = 256 threads, owns a 16(t) x 16(s) tile; D processed in 256-wide
// chunks staged in LDS (row stride padded to 260 floats -> conflict-free).
constexpr int CH = 256;        // d-chunk
constexpr int LR = CH + 4;     // padded LDS row stride (floats)

__global__ void __launch_bounds__(256)
mlp_score_tanh(const float* __restrict__ WQ,  // [Bc*Tc, Dc]
               const float* __restrict__ UH,  // [Bc*Sc, Dc]
               const float* __restrict__ V,   // [Dc]
               float* __restrict__ out) {     // [Bc, Tc, Sc]
    __shared__ float swq[16 * LR];
    __shared__ float suh[16 * LR];
    __shared__ float sv[Dc];

    const int s_tile = blockIdx.x;
    const int t_tile = blockIdx.y;
    const int b      = blockIdx.z;
    const int tid    = threadIdx.x;

    const float4* wqg = (const float4*)(WQ + (size_t)(b * Tc + t_tile * 16) * Dc);
    const float4* uhg = (const float4*)(UH + (size_t)(b * Sc + s_tile * 16) * Dc);

    if (tid < Dc / 4) ((float4*)sv)[tid] = ((const float4*)V)[tid];

    const int tl = tid >> 4;    // local t row (0..15)
    const int sl = tid & 15;    // local s row (0..15)
    const float* wr = &swq[tl * LR];
    const float* ur = &suh[sl * LR];

    float acc = 0.0f;
    for (int ch = 0; ch < Dc; ch += CH) {
        // Stage this d-chunk of both 16-row tiles (4 float4 per thread per tile).
        const int c4base = ch >> 2;
#pragma unroll
        for (int idx = tid; idx < 16 * (CH / 4); idx += 256) {
            const int row = idx >> 6;      // CH/4 == 64
            const int c4  = idx & 63;
            *(float4*)&swq[row * LR + c4 * 4] = wqg[row * (Dc / 4) + c4base + c4];
            *(float4*)&suh[row * LR + c4 * 4] = uhg[row * (Dc / 4) + c4base + c4];
        }
        // Prefetch the next chunk into cache while we compute this one.
        if (ch + CH < Dc) {
            __builtin_prefetch((const float*)wqg + (size_t)(tid & 15) * Dc + ch + CH, 0, 0);
            __builtin_prefetch((const float*)uhg + (size_t)(tid & 15) * Dc + ch + CH, 0, 0);
        }
        __syncthreads();

#pragma unroll 4
        for (int d = 0; d < CH; d += 4) {
            const float4 w  = *(const float4*)(wr + d);
            const float4 u  = *(const float4*)(ur + d);
            const float4 vv = *(const float4*)(sv + ch + d);
            acc = fmaf(vv.x, fast_tanh(w.x + u.x), acc);
            acc = fmaf(vv.y, fast_tanh(w.y + u.y), acc);
            acc = fmaf(vv.z, fast_tanh(w.z + u.z), acc);
            acc = fmaf(vv.w, fast_tanh(w.w + u.w), acc);
        }
        __syncthreads();
    }

    out[((size_t)(b * Tc + t_tile * 16 + tl)) * Sc + s_tile * 16 + sl] = acc;
}

extern "C" void kernel_launch(void* const* d_in, const int* in_sizes, int n_in,
                              void* d_out, int out_size, void* d_ws, size_t ws_size,
                              hipStream_t stream) {
    const float* h_t = (const float*)d_in[0];  // [B,T,D]
    const float* h_s = (const float*)d_in[1];  // [B,S,D]
    const float* Wq  = (const float*)d_in[2];  // [D,D]
    const float* bq  = (const float*)d_in[3];  // [D]
    const float* Wc  = (const float*)d_in[4];  // [D,D]
    const float* v   = (const float*)d_in[5];  // [D]
    float* out = (float*)d_out;                // [B,T,S]

    float* wq_ws = (float*)d_ws;                          // B*T*D floats (4 MB)
    float* uh_ws = wq_ws + (size_t)Bc * Tc * Dc;          // B*S*D floats (4 MB)

    // Projections via WMMA f32: one wave per 16x16 tile.
    dim3 ggrid(Dc / 16, (Bc * Tc) / 16);
    proj_gemm_wmma<<<ggrid, dim3(32), 0, stream>>>(h_t, Wq, bq, wq_ws);
    dim3 ggrid2(Dc / 16, (Bc * Sc) / 16);
    proj_gemm_wmma<<<ggrid2, dim3(32), 0, stream>>>(h_s, Wc, nullptr, uh_ws);

    // Broadcast-tanh-reduce, TRANS-pipe bound.
    dim3 sgrid(Sc / 16, Tc / 16, Bc);
    mlp_score_tanh<<<sgrid, dim3(256), 0, stream>>>(wq_ws, uh_ws, v, out);
}